// SupervisedPredictor_17901423690326
// MI455X (gfx1250) — compile-verified
//
#include <hip/hip_runtime.h>
#include <cstdint>

#define N_   50000
#define FIN_ 64
#define H_   128
#define E_   800000
#define T_   8
#define P_   250000
#define D_   500000
#define G_   20000
#define V_   2000
#define O_   2
#define H3_  384

typedef __attribute__((ext_vector_type(2))) float v2f;
typedef __attribute__((ext_vector_type(8))) float v8f;

__device__ __forceinline__ v8f wmma4(v2f a, v2f b, v8f c) {
  // V_WMMA_F32_16X16X4_F32 : D = A(16x4) * B(4x16) + C(16x16), all f32
  return __builtin_amdgcn_wmma_f32_16x16x4_f32(
      /*neg_a=*/false, a, /*neg_b=*/false, b,
      /*c_mod=*/(short)0, c, /*reuse_a=*/false, /*reuse_b=*/false);
}

__device__ __forceinline__ v2f ld2(const float* __restrict__ p) {
  return *(const v2f*)p;   // 8B-aligned by construction (even k offsets, 16B rows)
}

// ---------------------------------------------------------------- utilities
__global__ void k_zero(float* __restrict__ p, long n) {
  long i = (long)blockIdx.x * blockDim.x + threadIdx.x;
  long s = (long)gridDim.x * blockDim.x;
  for (; i < n; i += s) p[i] = 0.f;
}

__global__ void k_deg(const int* __restrict__ edst, float* __restrict__ deg, int e) {
  int i = blockIdx.x * blockDim.x + threadIdx.x;
  if (i < e) atomicAdd(&deg[edst[i]], 1.0f);
}

__global__ void k_invdeg(float* __restrict__ deg, int n) {  // in-place deg -> 1/max(deg,1)
  int i = blockIdx.x * blockDim.x + threadIdx.x;
  if (i < n) deg[i] = 1.0f / fmaxf(deg[i], 1.0f);
}

// W[k*Ncol + c]  ->  WT[c*K + k]
__global__ void k_transpose(const float* __restrict__ W, float* __restrict__ WT,
                            int K, int Ncol) {
  int idx = blockIdx.x * blockDim.x + threadIdx.x;
  if (idx < K * Ncol) {
    int k = idx / Ncol, c = idx % Ncol;
    WT[(long)c * K + k] = W[idx];
  }
}

// ----------------------------------------------- edge mean-agg (numerator)
template <int K>
__global__ void k_edge_agg(const float* __restrict__ h, const int* __restrict__ es,
                           const int* __restrict__ ed, float* __restrict__ agg) {
  const int CH = K / 4;
  long tid = (long)blockIdx.x * blockDim.x + threadIdx.x;
  if (tid >= (long)E_ * CH) return;
  int e = (int)(tid / CH);
  int c = (int)(tid % CH) * 4;
  int s = es[e], d = ed[e];
  const float4 v = *(const float4*)(h + (long)s * K + c);
  float* ap = agg + (long)d * K + c;
  atomicAdd(ap + 0, v.x); atomicAdd(ap + 1, v.y);
  atomicAdd(ap + 2, v.z); atomicAdd(ap + 3, v.w);
}

// ---- fused MP GEMM: relu(A@Ws + (agg*invdeg)@Wn + b), WsT/WnT transposed [c][k]
// All B-fragment loads: one base register + compile-time immediate offsets.
template <int K>
__global__ void k_gemm_mp(const float* __restrict__ A, const float* __restrict__ agg,
                          const float* __restrict__ invdeg,
                          const float* __restrict__ WsT, const float* __restrict__ WnT,
                          const float* __restrict__ bias, float* __restrict__ out,
                          int mtiles) {
  int wave = threadIdx.x >> 5, lane = threadIdx.x & 31;
  int tile = blockIdx.x * 8 + wave;
  if (tile >= mtiles) return;
  int half = lane >> 4, ml = lane & 15;
  int r = tile * 16 + ml;                 // A-fragment row for this lane
  float idg = invdeg[r];
  v8f acc[8];
#pragma unroll
  for (int nt = 0; nt < 8; ++nt) {
    float bv = bias[nt * 16 + ml];        // C column for this lane = ml
#pragma unroll
    for (int j = 0; j < 8; ++j) acc[nt][j] = bv;
  }
  const float* ab = A    + (long)r * K + half * 2;
  const float* gb = agg  + (long)r * K + half * 2;
  const float* sb = WsT  + (long)ml * K + half * 2;  // tile nt at +nt*16*K (immediate)
  const float* nb = WnT  + (long)ml * K + half * 2;
#pragma unroll 2
  for (int k0 = 0; k0 < K; k0 += 4) {
    v2f a1 = ld2(ab + k0);
    v2f g2 = ld2(gb + k0);
    v2f a2; a2[0] = g2[0] * idg; a2[1] = g2[1] * idg;
    v2f bs[8], bn[8];
#pragma unroll
    for (int nt = 0; nt < 8; ++nt) {
      bs[nt] = ld2(sb + nt * 16 * K + k0);
      bn[nt] = ld2(nb + nt * 16 * K + k0);
    }
#pragma unroll
    for (int nt = 0; nt < 8; ++nt) {
      acc[nt] = wmma4(a1, bs[nt], acc[nt]);
      acc[nt] = wmma4(a2, bn[nt], acc[nt]);
    }
  }
#pragma unroll
  for (int nt = 0; nt < 8; ++nt)
#pragma unroll
    for (int j = 0; j < 8; ++j) {
      int row = tile * 16 + half * 8 + j; // C rows: half selects 0-7 / 8-15
      out[(long)row * H_ + nt * 16 + ml] = fmaxf(acc[nt][j], 0.f);
    }
}

// ------------------------------------------------ dst-set segment sum [P,H]
__global__ void k_dstf(const float* __restrict__ feats, const int* __restrict__ dn,
                       const int* __restrict__ ds, const int* __restrict__ tix,
                       float* __restrict__ dstf) {
  const int CH = H_ / 4;
  long tid = (long)blockIdx.x * blockDim.x + threadIdx.x;
  if (tid >= (long)D_ * CH) return;
  int d = (int)(tid / CH);
  int c = (int)(tid % CH) * 4;
  int node = dn[d], seg = ds[d], t = tix[seg];
  const float4 v = *(const float4*)(feats + ((long)t * N_ + node) * H_ + c);
  float* p = dstf + (long)seg * H_ + c;
  atomicAdd(p + 0, v.x); atomicAdd(p + 1, v.y);
  atomicAdd(p + 2, v.z); atomicAdd(p + 3, v.w);
}

// ---- fused decoder: gather cat -> relu(cat@Wd1+bd1)@Wd2+bd2 -> log_softmax
// Wd1T transposed [c][k]; 2 passes x 12 column tiles; 3 k-segment sub-loops so
// every load is base-pointer + immediate offset.
__global__ void k_decoder(const float* __restrict__ feats, const float* __restrict__ dstf,
                          const int* __restrict__ tix, const int* __restrict__ pix,
                          const int* __restrict__ six,
                          const float* __restrict__ Wd1T, const float* __restrict__ bd1,
                          const float* __restrict__ Wd2, const float* __restrict__ bd2,
                          float* __restrict__ logprob, int mtiles) {
  int wave = threadIdx.x >> 5, lane = threadIdx.x & 31;
  int tile = blockIdx.x * 8 + wave;
  if (tile >= mtiles) return;
  int half = lane >> 4, ml = lane & 15;
  int r = tile * 16 + ml;
  int ti = tix[r], pi = pix[r], si = six[r];
  const float* aseg0 = feats + ((long)ti * N_ + pi) * H_ + half * 2;
  const float* aseg1 = feats + ((long)ti * N_ + si) * H_ + half * 2;
  const float* aseg2 = dstf + (long)r * H_ + half * 2;

  float pl0[8], pl1[8];
#pragma unroll
  for (int j = 0; j < 8; ++j) { pl0[j] = 0.f; pl1[j] = 0.f; }

  for (int pass = 0; pass < 2; ++pass) {
    v8f acc[12];
#pragma unroll
    for (int nt = 0; nt < 12; ++nt) {
      float bv = bd1[(pass * 12 + nt) * 16 + ml];
#pragma unroll
      for (int j = 0; j < 8; ++j) acc[nt][j] = bv;
    }
    // B base for this pass: column (pass*12*16 + ml); tile nt at +nt*16*H3 (imm)
    const float* bb0 = Wd1T + ((long)pass * 12 * 16 + ml) * H3_ + half * 2;
#pragma unroll
    for (int seg = 0; seg < 3; ++seg) {
      const float* ab = (seg == 0) ? aseg0 : (seg == 1) ? aseg1 : aseg2;
      const float* bb = bb0 + seg * H_;
#pragma unroll 2
      for (int k0 = 0; k0 < H_; k0 += 4) {
        v2f a = ld2(ab + k0);
        v2f bf[12];
#pragma unroll
        for (int nt = 0; nt < 12; ++nt) bf[nt] = ld2(bb + nt * 16 * H3_ + k0);
#pragma unroll
        for (int nt = 0; nt < 12; ++nt) acc[nt] = wmma4(a, bf[nt], acc[nt]);
      }
    }
    // fold this pass's columns into logits partials:
    // lane owns column c = (pass*12+nt)*16+ml, rows half*8 + j
#pragma unroll
    for (int nt = 0; nt < 12; ++nt) {
      int c = (pass * 12 + nt) * 16 + ml;
      float w0 = Wd2[c * 2 + 0], w1 = Wd2[c * 2 + 1];
#pragma unroll
      for (int j = 0; j < 8; ++j) {
        float hrel = fmaxf(acc[nt][j], 0.f);
        pl0[j] += hrel * w0; pl1[j] += hrel * w1;
      }
    }
  }
  // butterfly-reduce across the 16 lanes of each half (masks stay in-half)
#pragma unroll
  for (int j = 0; j < 8; ++j)
    for (int m = 1; m < 16; m <<= 1) {
      pl0[j] += __shfl_xor(pl0[j], m, 32);
      pl1[j] += __shfl_xor(pl1[j], m, 32);
    }
  if (ml == 0) {
    float b0 = bd2[0], b1 = bd2[1];
    for (int j = 0; j < 8; ++j) {
      int row = tile * 16 + half * 8 + j;
      float l0 = pl0[j] + b0, l1 = pl1[j] + b1;
      float mx = fmaxf(l0, l1);
      float lse = mx + logf(expf(l0 - mx) + expf(l1 - mx));
      logprob[(long)row * 2 + 0] = l0 - lse;
      logprob[(long)row * 2 + 1] = l1 - lse;
    }
  }
}

// -------------------------------------------------- epilogue (tiny) kernels
__global__ void k_seg2(const float* __restrict__ in, const int* __restrict__ seg,
                       float* __restrict__ out, int n) {
  int i = blockIdx.x * blockDim.x + threadIdx.x;
  if (i < n) {
    int g = seg[i];
    atomicAdd(&out[(long)g * 2 + 0], in[(long)i * 2 + 0]);
    atomicAdd(&out[(long)g * 2 + 1], in[(long)i * 2 + 1]);
  }
}

__global__ void k_scan2(const float* __restrict__ in, float* __restrict__ out, int rows) {
  __shared__ float s0[256], s1[256];
  int tid = threadIdx.x;
  float c0 = 0.f, c1 = 0.f;
  for (int base = 0; base < rows; base += 256) {
    int i = base + tid;
    s0[tid] = (i < rows) ? in[(long)i * 2 + 0] : 0.f;
    s1[tid] = (i < rows) ? in[(long)i * 2 + 1] : 0.f;
    __syncthreads();
    for (int off = 1; off < 256; off <<= 1) {
      float t0 = (tid >= off) ? s0[tid - off] : 0.f;
      float t1 = (tid >= off) ? s1[tid - off] : 0.f;
      __syncthreads();
      s0[tid] += t0; s1[tid] += t1;
      __syncthreads();
    }
    if (i < rows) {
      out[(long)i * 2 + 0] = s0[tid] + c0;
      out[(long)i * 2 + 1] = s1[tid] + c1;
    }
    __syncthreads();
    c0 += s0[255]; c1 += s1[255];
    __syncthreads();
  }
}

__global__ void k_countsum(const float* __restrict__ counts, float* __restrict__ tot) {
  __shared__ float s[256];
  float v = 0.f;
  for (int i = threadIdx.x; i < V_; i += 256) v += counts[i];
  s[threadIdx.x] = v; __syncthreads();
  for (int off = 128; off > 0; off >>= 1) {
    if (threadIdx.x < off) s[threadIdx.x] += s[threadIdx.x + off];
    __syncthreads();
  }
  if (threadIdx.x == 0) *tot = s[0];
}

__global__ void k_logc(const float* __restrict__ counts, const float* __restrict__ tot,
                       float* __restrict__ logc) {
  int i = blockIdx.x * blockDim.x + threadIdx.x;
  if (i < V_) logc[i] = logf(counts[i]) - logf(*tot);
}

__global__ void k_addG(const float* __restrict__ cumG, const float* __restrict__ cumV,
                       const float* __restrict__ vtot, const float* __restrict__ logc,
                       const int* __restrict__ vog, float* __restrict__ addG) {
  int g = blockIdx.x * blockDim.x + threadIdx.x;
  if (g < G_) {
    int vg = vog[g];
    float lc = logc[vg];
#pragma unroll
    for (int o = 0; o < 2; ++o) {
      float within = cumG[(long)g * 2 + o] - (cumV[(long)vg * 2 + o] - vtot[(long)vg * 2 + o]);
      addG[(long)g * 2 + o] = within + lc;
    }
  }
}

__global__ void k_final(const float* __restrict__ logprob, const float* __restrict__ addG,
                        const int* __restrict__ gid, const int* __restrict__ pix,
                        float* __restrict__ out) {
  int p = blockIdx.x * blockDim.x + threadIdx.x;
  if (p < P_) {
    int g = gid[p], pl = pix[p];
    atomicAdd(&out[(long)pl * 2 + 0], logprob[(long)p * 2 + 0] + addG[(long)g * 2 + 0]);
    atomicAdd(&out[(long)pl * 2 + 1], logprob[(long)p * 2 + 1] + addG[(long)g * 2 + 1]);
  }
}

// ------------------------------------------------------------------- driver
extern "C" void kernel_launch(void* const* d_in, const int* in_sizes, int n_in,
                              void* d_out, int out_size, void* d_ws, size_t ws_size,
                              hipStream_t stream) {
  const float* x     = (const float*)d_in[0];
  const float* Wes   = (const float*)d_in[1];
  const float* Wen   = (const float*)d_in[2];
  const float* benc  = (const float*)d_in[3];
  const float* W2s   = (const float*)d_in[4];
  const float* W2n   = (const float*)d_in[5];
  const float* b2    = (const float*)d_in[6];
  const float* Wd1   = (const float*)d_in[7];
  const float* bd1   = (const float*)d_in[8];
  const float* Wd2   = (const float*)d_in[9];
  const float* bd2   = (const float*)d_in[10];
  const float* cnts  = (const float*)d_in[11];
  const int* esrc    = (const int*)d_in[12];
  const int* edst    = (const int*)d_in[13];
  const int* pix     = (const int*)d_in[14];
  const int* six     = (const int*)d_in[15];
  const int* tix     = (const int*)d_in[16];
  const int* dnodes  = (const int*)d_in[17];
  const int* dseg    = (const int*)d_in[18];
  const int* gid     = (const int*)d_in[19];
  const int* vog     = (const int*)d_in[20];
  float* out = (float*)d_out;

  // workspace carving (256B aligned)
  char* w = (char*)d_ws;
  auto carve = [&](size_t bytes) -> float* {
    float* p = (float*)w;
    w += (bytes + 255) & ~(size_t)255;
    return p;
  };
  float* feats  = carve((size_t)9 * N_ * H_ * 4);   // [T+1, N, H]
  float* agg    = carve((size_t)N_ * H_ * 4);
  float* invdeg = carve((size_t)N_ * 4);
  float* dstf   = carve((size_t)P_ * H_ * 4);
  float* lgp    = carve((size_t)P_ * 2 * 4);
  float* gsum   = carve((size_t)G_ * 2 * 4);
  float* cumG   = carve((size_t)G_ * 2 * 4);
  float* vtot   = carve((size_t)V_ * 2 * 4);
  float* cumV   = carve((size_t)V_ * 2 * 4);
  float* addG   = carve((size_t)G_ * 2 * 4);
  float* logc   = carve((size_t)V_ * 4);
  float* tot    = carve(4);
  float* WesT   = carve((size_t)FIN_ * H_ * 4);     // transposed weights [c][k]
  float* WenT   = carve((size_t)FIN_ * H_ * 4);
  float* W2sT   = carve((size_t)H_ * H_ * 4);
  float* W2nT   = carve((size_t)H_ * H_ * 4);
  float* Wd1T   = carve((size_t)H3_ * H3_ * 4);

  auto zero = [&](float* p, long n) { k_zero<<<2048, 256, 0, stream>>>(p, n); };

  // weight transposes (one-time per call; tiny)
  k_transpose<<<(FIN_ * H_ + 255) / 256, 256, 0, stream>>>(Wes, WesT, FIN_, H_);
  k_transpose<<<(FIN_ * H_ + 255) / 256, 256, 0, stream>>>(Wen, WenT, FIN_, H_);
  k_transpose<<<(H_ * H_ + 255) / 256, 256, 0, stream>>>(W2s, W2sT, H_, H_);
  k_transpose<<<(H_ * H_ + 255) / 256, 256, 0, stream>>>(W2n, W2nT, H_, H_);
  k_transpose<<<(H3_ * H3_ + 255) / 256, 256, 0, stream>>>(Wd1, Wd1T, H3_, H3_);

  // degree -> invdeg
  zero(invdeg, N_);
  k_deg<<<(E_ + 255) / 256, 256, 0, stream>>>(edst, invdeg, E_);
  k_invdeg<<<(N_ + 255) / 256, 256, 0, stream>>>(invdeg, N_);

  const int MT_N = N_ / 16;   // 3125
  const int MT_P = P_ / 16;   // 15625

  // encoder: feats[0] = relu(x@Wes + mean@Wen + benc)
  zero(agg, (long)N_ * FIN_);
  {
    long tot_thr = (long)E_ * (FIN_ / 4);
    k_edge_agg<FIN_><<<(unsigned)((tot_thr + 255) / 256), 256, 0, stream>>>(x, esrc, edst, agg);
  }
  k_gemm_mp<FIN_><<<(MT_N + 7) / 8, 256, 0, stream>>>(x, agg, invdeg, WesT, WenT, benc,
                                                      feats, MT_N);

  // T sequential MP steps
  for (int t = 0; t < T_; ++t) {
    const float* hin = feats + (size_t)t * N_ * H_;
    float* hout = feats + (size_t)(t + 1) * N_ * H_;
    zero(agg, (long)N_ * H_);
    long tot_thr = (long)E_ * (H_ / 4);
    k_edge_agg<H_><<<(unsigned)((tot_thr + 255) / 256), 256, 0, stream>>>(hin, esrc, edst, agg);
    k_gemm_mp<H_><<<(MT_N + 7) / 8, 256, 0, stream>>>(hin, agg, invdeg, W2sT, W2nT, b2,
                                                      hout, MT_N);
  }

  // dst segment features
  zero(dstf, (long)P_ * H_);
  {
    long tot_thr = (long)D_ * (H_ / 4);
    k_dstf<<<(unsigned)((tot_thr + 255) / 256), 256, 0, stream>>>(feats, dnodes, dseg, tix, dstf);
  }

  // fused decoder -> logprob
  k_decoder<<<(MT_P + 7) / 8, 256, 0, stream>>>(feats, dstf, tix, pix, six,
                                                Wd1T, bd1, Wd2, bd2, lgp, MT_P);

  // group / variant bookkeeping
  zero(gsum, (long)G_ * 2);
  k_seg2<<<(P_ + 255) / 256, 256, 0, stream>>>(lgp, gid, gsum, P_);
  k_scan2<<<1, 256, 0, stream>>>(gsum, cumG, G_);
  zero(vtot, (long)V_ * 2);
  k_seg2<<<(G_ + 255) / 256, 256, 0, stream>>>(gsum, vog, vtot, G_);
  k_scan2<<<1, 256, 0, stream>>>(vtot, cumV, V_);
  k_countsum<<<1, 256, 0, stream>>>(cnts, tot);
  k_logc<<<(V_ + 255) / 256, 256, 0, stream>>>(cnts, tot, logc);
  k_addG<<<(G_ + 255) / 256, 256, 0, stream>>>(cumG, cumV, vtot, logc, vog, addG);

  // final scatter-add into output
  zero(out, (long)N_ * 2);
  k_final<<<(P_ + 255) / 256, 256, 0, stream>>>(lgp, addG, gid, pix, out);
}